// gGRUCell_11252814315782
// MI455X (gfx1250) — compile-verified
//
#include <hip/hip_runtime.h>
#include <math.h>

typedef __attribute__((ext_vector_type(2))) float v2f;
typedef __attribute__((ext_vector_type(8))) float v8f;

constexpr int Nn = 16;    // batch
constexpr int Vv = 4096;  // positions
constexpr int Ii = 128;   // input features
constexpr int Oo = 128;   // output features
constexpr int Gg = 16;    // groups

// One wave32 handles one v and one 64-wide half of O.
// Per wave: 3 gates x 4 tiles of 16x16 f32 accumulators = 96 VGPRs.
__global__ __launch_bounds__(128, 1) void ggru_cell_kernel(
    const float* __restrict__ x,    // [N,V,I]
    const float* __restrict__ hx,   // [N,V,O]
    const int*   __restrict__ gidx, // [V]
    const float* __restrict__ wih,  // [3,G,O,I]
    const float* __restrict__ bih,  // [3,G,O]
    const float* __restrict__ whh,  // [3,G,O,O] (only diagonal used)
    const float* __restrict__ bhh,  // [3,G,O]
    float*       __restrict__ out)  // [N,V,O]
{
    const int wave  = blockIdx.x * (blockDim.x >> 5) + (threadIdx.x >> 5);
    const int v     = wave >> 1;       // 0..4095
    const int ohalf = wave & 1;        // which 64-wide half of O
    const int lane  = threadIdx.x & 31;
    const int row16 = lane & 15;       // M (for A) / N (for B,C) within a tile
    const int khalf = lane >> 4;       // 0: K={0,1}, 1: K={2,3} per ISA 16x4 f32 layout

    const int g     = gidx[v];
    const int obase = ohalf * 64;

    // A fragment source: x[n=row16, v, k + 2*khalf + {0,1}]  -> 8-byte load
    const float* Aptr = x + (size_t)row16 * (Vv * Ii) + (size_t)v * Ii + 2 * khalf;
    // B fragment source: B[k,o] = w[o,k]; lane holds row o = obase + t*16 + row16,
    // consecutive i = k + 2*khalf + {0,1} -> 8-byte load along the weight row.
    const float* B0 = wih + ((size_t)(0 * Gg + g) * Oo + obase + row16) * Ii + 2 * khalf;
    const float* B1 = wih + ((size_t)(1 * Gg + g) * Oo + obase + row16) * Ii + 2 * khalf;
    const float* B2 = wih + ((size_t)(2 * Gg + g) * Oo + obase + row16) * Ii + 2 * khalf;

    v8f acc[3][4];
#pragma unroll
    for (int gt = 0; gt < 3; ++gt)
#pragma unroll
        for (int t = 0; t < 4; ++t)
            acc[gt][t] = (v8f)0.0f;

    // K loop: 32 steps of K=4, 12 WMMAs per step (3 gates x 4 O-tiles)
    for (int k = 0; k < Ii; k += 4) {
        v2f a = *reinterpret_cast<const v2f*>(Aptr + k);
#pragma unroll
        for (int t = 0; t < 4; ++t) {
            v2f b0 = *reinterpret_cast<const v2f*>(B0 + (size_t)(t * 16) * Ii + k);
            v2f b1 = *reinterpret_cast<const v2f*>(B1 + (size_t)(t * 16) * Ii + k);
            v2f b2 = *reinterpret_cast<const v2f*>(B2 + (size_t)(t * 16) * Ii + k);
            acc[0][t] = __builtin_amdgcn_wmma_f32_16x16x4_f32(
                false, a, false, b0, (short)0, acc[0][t], false, false);
            acc[1][t] = __builtin_amdgcn_wmma_f32_16x16x4_f32(
                false, a, false, b1, (short)0, acc[1][t], false, false);
            acc[2][t] = __builtin_amdgcn_wmma_f32_16x16x4_f32(
                false, a, false, b2, (short)0, acc[2][t], false, false);
        }
    }

    // Fused GRU epilogue. C/D layout: VGPR r, lane L -> M = r + 8*(L>>4), N = L&15.
#pragma unroll
    for (int t = 0; t < 4; ++t) {
        const int o = obase + t * 16 + row16;

        const float bi0 = bih[(0 * Gg + g) * Oo + o];
        const float bi1 = bih[(1 * Gg + g) * Oo + o];
        const float bi2 = bih[(2 * Gg + g) * Oo + o];
        const float bh0 = bhh[(0 * Gg + g) * Oo + o];
        const float bh1 = bhh[(1 * Gg + g) * Oo + o];
        const float bh2 = bhh[(2 * Gg + g) * Oo + o];
        const float wd0 = whh[((size_t)(0 * Gg + g) * Oo + o) * Oo + o];
        const float wd1 = whh[((size_t)(1 * Gg + g) * Oo + o) * Oo + o];
        const float wd2 = whh[((size_t)(2 * Gg + g) * Oo + o) * Oo + o];

#pragma unroll
        for (int r = 0; r < 8; ++r) {
            const int n = r + 8 * khalf;
            const size_t off = (size_t)n * (Vv * Oo) + (size_t)v * Oo + o;
            const float h  = hx[off];
            const float pr = acc[0][t][r] + bi0 + h * wd0 + bh0;
            const float pz = acc[1][t][r] + bi1 + h * wd1 + bh1;
            const float rg = 1.0f / (1.0f + __expf(-pr));
            const float zg = 1.0f / (1.0f + __expf(-pz));
            const float pn = acc[2][t][r] + bi2 + rg * (h * wd2 + bh2);
            const float ng = tanhf(pn);
            out[off] = (1.0f - zg) * ng + zg * h;
        }
    }
}

extern "C" void kernel_launch(void* const* d_in, const int* in_sizes, int n_in,
                              void* d_out, int out_size, void* d_ws, size_t ws_size,
                              hipStream_t stream) {
    const float* x   = (const float*)d_in[0];
    const float* hx  = (const float*)d_in[1];
    const int*   gi  = (const int*)d_in[2];
    const float* wih = (const float*)d_in[3];
    const float* bih = (const float*)d_in[4];
    const float* whh = (const float*)d_in[5];
    const float* bhh = (const float*)d_in[6];
    float* out = (float*)d_out;

    // V*2 waves total, 4 waves (128 threads) per block -> 2048 blocks.
    dim3 block(128);
    dim3 grid((Vv * 2) / 4);
    ggru_cell_kernel<<<grid, block, 0, stream>>>(x, hx, gi, wih, bih, whh, bhh, out);
}